// BMGAE_81810537054267
// MI455X (gfx1250) — compile-verified
//
#include <hip/hip_runtime.h>

// ---------------------------------------------------------------------------
// BMGAE LightGCN propagate for MI455X (gfx1250).
//  - scatter_edges: wave32 per edge; edge metadata scalarized (s_load path),
//    float2/lane coalesced feature reads, global_atomic_add_f32 scatter into
//    L2-resident accumulators.
//  - finish_norm_wmma: 16-row tile per wave; row sum-of-squares computed as
//    (X.^2) @ ones via v_wmma_f32_16x16x4_f32 (16-step K accumulation),
//    then out = base + scale*raw / max(scale*sqrt(ss), 1e-12).
// ---------------------------------------------------------------------------

typedef __attribute__((ext_vector_type(2))) float v2f;
typedef __attribute__((ext_vector_type(8))) float v8f;

// One wave (32 lanes) per edge; lane l handles feature dims [2l, 2l+1].
// Edge index is wave-uniform and scalarized so src/dst/val lower to SMEM
// loads (scalar cache, KMcnt) instead of broadcast vector loads.
__global__ __launch_bounds__(256)
void scatter_edges(const int* __restrict__ src, const int* __restrict__ dst,
                   const float* __restrict__ val,
                   const float* __restrict__ p1, const float* __restrict__ p2,
                   int n_left, float* __restrict__ nxt, int nE) {
    const int lane = threadIdx.x & 31;
    const int wib  = __builtin_amdgcn_readfirstlane(threadIdx.x >> 5);
    const int wid  = blockIdx.x * (blockDim.x >> 5) + wib;
    const int nw   = gridDim.x * (blockDim.x >> 5);
    for (int e = wid; e < nE; e += nw) {
        const int   s = src[e];     // scalar index -> s_load_b32
        const int   d = dst[e];
        const float w = val[e];
        const float* fp = (d < n_left) ? (p1 + (size_t)d * 64)
                                       : (p2 + (size_t)(d - n_left) * 64);
        const float2 x = *(const float2*)(fp + lane * 2);
        float* q = nxt + (size_t)s * 64 + lane * 2;
        atomicAdd(q,     w * x.x);
        atomicAdd(q + 1, w * x.y);
    }
}

// One wave per 16-row tile (n_rows multiple of 16 for this kernel).
// LDS row stride 68 floats -> bank-conflict-free WMMA-layout reads.
__global__ __launch_bounds__(32)
void finish_norm_wmma(const float* __restrict__ raw, float scale,
                      const float* __restrict__ base1, const float* __restrict__ base2,
                      int n_left, float* __restrict__ out1, float* __restrict__ out2,
                      int accumulate, int n_rows) {
    __shared__ float tile[16 * 68];
    const int lane  = threadIdx.x;        // 0..31
    const int tile0 = blockIdx.x * 16;
    if (tile0 >= n_rows) return;          // uniform; EXEC stays all-1s below

    // Stage 16x64 tile, coalesced b128 loads.
    #pragma unroll
    for (int p = 0; p < 8; ++p) {
        const int id = p * 32 + lane;     // float4 index within tile
        const int r  = id >> 4;
        const int c4 = id & 15;
        const float4 v = *(const float4*)(raw + (size_t)(tile0 + r) * 64 + c4 * 4);
        *(float4*)&tile[r * 68 + c4 * 4] = v;
    }
    __syncthreads();

    // sumsq[m] = (X.^2 @ ones) via 16x16x4 f32 WMMA accumulation.
    // A layout: lane m / m+16 holds row m, K = {4k,4k+1} / {4k+2,4k+3}.
    const int m    = lane & 15;
    const int half = lane >> 4;
    const v2f ones = {1.0f, 1.0f};
    v8f c = {};
    #pragma unroll
    for (int k = 0; k < 16; ++k) {
        const int col = 4 * k + 2 * half;
        v2f a;
        a.x = tile[m * 68 + col];
        a.y = tile[m * 68 + col + 1];
        const v2f a2 = a * a;
        c = __builtin_amdgcn_wmma_f32_16x16x4_f32(
                false, a2, false, ones, (short)0, c, false, false);
    }

    // C VGPR r holds sumsq[r + 8*half] replicated across N -> static extract.
    #pragma unroll
    for (int r = 0; r < 8; ++r) {
        const int   row   = r + 8 * half;            // 0..15 within tile
        const float ss    = c[r];
        const float denom = fmaxf(scale * sqrtf(ss), 1e-12f);
        const float inv   = scale / denom;
        const int   g     = tile0 + row;
        float*       op;
        const float* bp;
        if (g < n_left) { op = out1 + (size_t)g * 64;            bp = base1 + (size_t)g * 64; }
        else            { op = out2 + (size_t)(g - n_left) * 64; bp = base2 + (size_t)(g - n_left) * 64; }
        const int col = lane & 15;
        #pragma unroll
        for (int cc = 0; cc < 4; ++cc) {
            const int cidx = col + 16 * cc;
            float v = tile[row * 68 + cidx] * inv;
            v += accumulate ? op[cidx] : bp[cidx];
            op[cidx] = v;
        }
    }
}

// Shuffle-reduction fallback for a row-range tail (n_rows % 16 != 0).
__global__ __launch_bounds__(32)
void finish_norm_tail(const float* __restrict__ raw, float scale,
                      const float* __restrict__ base1, const float* __restrict__ base2,
                      int n_left, float* __restrict__ out1, float* __restrict__ out2,
                      int accumulate, int row0, int n_rows) {
    const int row = row0 + blockIdx.x;
    if (row >= n_rows) return;
    const int lane = threadIdx.x;
    const float2 x = *(const float2*)(raw + (size_t)row * 64 + lane * 2);
    float ss = x.x * x.x + x.y * x.y;
    #pragma unroll
    for (int o = 16; o > 0; o >>= 1) ss += __shfl_xor(ss, o, 32);
    const float denom = fmaxf(scale * sqrtf(ss), 1e-12f);
    const float inv   = scale / denom;
    float*       op;
    const float* bp;
    if (row < n_left) { op = out1 + (size_t)row * 64;            bp = base1 + (size_t)row * 64; }
    else              { op = out2 + (size_t)(row - n_left) * 64; bp = base2 + (size_t)(row - n_left) * 64; }
    float2 o2;
    o2.x = x.x * inv;
    o2.y = x.y * inv;
    const float2 addv = accumulate ? *(const float2*)(op + lane * 2)
                                   : *(const float2*)(bp + lane * 2);
    o2.x += addv.x;
    o2.y += addv.y;
    *(float2*)(op + lane * 2) = o2;
}

static void run_graph(const int* src, const int* dst, const float* val, int nE,
                      const float* rep1, const float* rep2, int nl, int nr,
                      float* out1, float* out2, float* next1, float* next2,
                      hipStream_t stream) {
    const int n = nl + nr;
    const int nTiles = n / 16;
    const int rem0   = nTiles * 16;
    const int remCnt = n - rem0;
    const int sBlocks = (nE + 7) / 8;

    // Layer 1: raw1 = A @ reps0 ; out = reps0 + l2norm(raw1/2)
    hipMemsetAsync(next1, 0, (size_t)n * 64 * sizeof(float), stream);
    scatter_edges<<<sBlocks, 256, 0, stream>>>(src, dst, val, rep1, rep2, nl, next1, nE);
    if (nTiles > 0)
        finish_norm_wmma<<<nTiles, 32, 0, stream>>>(next1, 0.5f, rep1, rep2, nl,
                                                    out1, out2, 0, n);
    if (remCnt > 0)
        finish_norm_tail<<<remCnt, 32, 0, stream>>>(next1, 0.5f, rep1, rep2, nl,
                                                    out1, out2, 0, rem0, n);

    // Layer 2: raw2 = A @ raw1 = 6 * cur2 ; out += l2norm(raw2/6)
    // (l2norm is scale-invariant, so layer-1 raw sums feed layer 2 directly
    //  and the 1/2*1/3 scaling is folded into the norm step's `scale`.)
    hipMemsetAsync(next2, 0, (size_t)n * 64 * sizeof(float), stream);
    scatter_edges<<<sBlocks, 256, 0, stream>>>(src, dst, val, next1,
                                               next1 + (size_t)nl * 64, nl, next2, nE);
    const float s2 = 1.0f / 6.0f;
    if (nTiles > 0)
        finish_norm_wmma<<<nTiles, 32, 0, stream>>>(next2, s2, rep1, rep2, nl,
                                                    out1, out2, 1, n);
    if (remCnt > 0)
        finish_norm_tail<<<remCnt, 32, 0, stream>>>(next2, s2, rep1, rep2, nl,
                                                    out1, out2, 1, rem0, n);
}

extern "C" void kernel_launch(void* const* d_in, const int* in_sizes, int n_in,
                              void* d_out, int out_size, void* d_ws, size_t ws_size,
                              hipStream_t stream) {
    const float* users   = (const float*)d_in[0];
    const float* bundles = (const float*)d_in[1];
    const float* items   = (const float*)d_in[2];
    const int*   ui_src  = (const int*)d_in[3];
    const int*   ui_dst  = (const int*)d_in[4];
    const float* ui_val  = (const float*)d_in[5];
    const int*   ub_src  = (const int*)d_in[6];
    const int*   ub_dst  = (const int*)d_in[7];
    const float* ub_val  = (const float*)d_in[8];
    const int*   bi_src  = (const int*)d_in[9];
    const int*   bi_dst  = (const int*)d_in[10];
    const float* bi_val  = (const float*)d_in[11];

    const int NU = in_sizes[0] / 64;
    const int NB = in_sizes[1] / 64;
    const int NI = in_sizes[2] / 64;
    const int nUI = in_sizes[3];
    const int nUB = in_sizes[6];
    const int nBI = in_sizes[9];

    size_t nmax = (size_t)(NU + NI);
    if ((size_t)(NU + NB) > nmax) nmax = (size_t)(NU + NB);
    if ((size_t)(NB + NI) > nmax) nmax = (size_t)(NB + NI);

    float* next1 = (float*)d_ws;
    float* next2 = next1 + nmax * 64;

    float* out = (float*)d_out;
    float* il_u = out;
    float* bl_u = out + (size_t)NU * 64;
    float* bl_b = out + (size_t)(2 * NU) * 64;
    float* bs_b = out + (size_t)(2 * NU + NB) * 64;
    float* il_i = out + (size_t)(2 * NU + 2 * NB) * 64;
    float* bs_i = out + (size_t)(2 * NU + 2 * NB + NI) * 64;

    run_graph(ui_src, ui_dst, ui_val, nUI, users,   items,   NU, NI, il_u, il_i,
              next1, next2, stream);
    run_graph(ub_src, ub_dst, ub_val, nUB, users,   bundles, NU, NB, bl_u, bl_b,
              next1, next2, stream);
    run_graph(bi_src, bi_dst, bi_val, nBI, bundles, items,   NB, NI, bs_b, bs_i,
              next1, next2, stream);
}